// TransitionDown_61478161875081
// MI455X (gfx1250) — compile-verified
//
#include <hip/hip_runtime.h>

// Problem constants (from reference): B=8, N=4096, C_in=128, C_out=256, K=16, S=2048
#define BB     8
#define NPTS   4096
#define CIN    128
#define COUT   256
#define KNN    16
#define SS_    2048
#define NROWS  (BB * SS_ * KNN)            // 262144 grouped rows
#define NQUADS ((BB * SS_ / 2) * (COUT / 32))  // 2x2 tiles per wave -> 65536

typedef float v2f __attribute__((ext_vector_type(2)));
typedef float v8f __attribute__((ext_vector_type(8)));

// ---------------------------------------------------------------------------
// Kernel 0: zero the stats accumulators (ws is poisoned, must re-zero per call)
// ---------------------------------------------------------------------------
__global__ void td_zero_kernel(float* stats) {
    stats[threadIdx.x] = 0.0f;   // 512 threads: sum[256] + sumsq[256]
}

// ---------------------------------------------------------------------------
// Kernel 1: KNN (squared distance == reference's -2<q,p>+|q|^2+|p|^2 ordering)
// One block = 256 queries of one batch; all 4096 points staged in 48KB LDS.
// Also writes the new_xyz tail of the output (xyz[:, :S]).
// ---------------------------------------------------------------------------
__global__ void td_knn_kernel(const float* __restrict__ xyz,
                              int* __restrict__ idxg,
                              float* __restrict__ out) {
    __shared__ float pts[NPTS * 3];   // 48 KB of the 320 KB WGP LDS
    const int tid = threadIdx.x;
    const int b   = blockIdx.x >> 3;               // 8 s-blocks per batch
    const int s   = ((blockIdx.x & 7) << 8) + tid; // 0..2047

    const float* src = xyz + (size_t)b * NPTS * 3;
    for (int i = tid; i < NPTS * 3; i += 256) pts[i] = src[i];
    __syncthreads();

    const float qx = pts[s * 3 + 0];
    const float qy = pts[s * 3 + 1];
    const float qz = pts[s * 3 + 2];

    float bd[KNN]; int bi[KNN];
#pragma unroll
    for (int k = 0; k < KNN; ++k) { bd[k] = 3.4e38f; bi[k] = 0; }

    for (int n = 0; n < NPTS; ++n) {
        const float dx = pts[n * 3 + 0] - qx;
        const float dy = pts[n * 3 + 1] - qy;
        const float dz = pts[n * 3 + 2] - qz;
        const float d  = dx * dx + dy * dy + dz * dz;
        if (d < bd[KNN - 1]) {
            int j = KNN - 1;
            while (j > 0 && bd[j - 1] > d) {
                bd[j] = bd[j - 1]; bi[j] = bi[j - 1]; --j;
            }
            bd[j] = d; bi[j] = n;
        }
    }

    const size_t gbase = ((size_t)b * SS_ + s) * KNN;
    for (int k = 0; k < KNN; ++k) idxg[gbase + k] = b * NPTS + bi[k];

    // new_xyz = xyz[:, :S]  (s < S <= N, coords already in registers)
    float* tail = out + (size_t)BB * SS_ * COUT;
    const size_t t = ((size_t)b * SS_ + s) * 3;
    tail[t + 0] = qx; tail[t + 1] = qy; tail[t + 2] = qz;
}

// ---------------------------------------------------------------------------
// Shared WMMA 2x2 register tile: two bs-tiles (each = one (b,s), k=0..15) x
// two 16-channel tiles, accumulated over C_in=128 with V_WMMA_F32_16X16X4_F32.
// Inner loop: 4 float2 loads -> 4 WMMAs (1 load per WMMA issue).
// fp32 A 16x4 layout: lane<16 -> M=lane, K=(kk,kk+1); lane>=16 -> M=lane-16,
// K=(kk+2,kk+3). B 4x16 mirrors it per-column.
// ---------------------------------------------------------------------------
__device__ __forceinline__ void td_gemm_tile22(const float* __restrict__ x,
                                               const float* __restrict__ W,
                                               const int* __restrict__ idxg,
                                               int bs0, int o0, int lid,
                                               v8f& acc00, v8f& acc01,
                                               v8f& acc10, v8f& acc11) {
    const int half = lid >> 4;      // K-half selector
    const int m    = lid & 15;      // row within tile / column within tile
    const int p0   = idxg[(bs0 + 0) * KNN + m];
    const int p1   = idxg[(bs0 + 1) * KNN + m];
    const float* ap0 = x + (size_t)p0 * CIN + 2 * half;
    const float* ap1 = x + (size_t)p1 * CIN + 2 * half;
    const float* bp0 = W + (size_t)(o0 + m) * CIN + 2 * half;
    const float* bp1 = W + (size_t)(o0 + 16 + m) * CIN + 2 * half;

    acc00 = (v8f){0.f,0.f,0.f,0.f,0.f,0.f,0.f,0.f};
    acc01 = acc00; acc10 = acc00; acc11 = acc00;
#pragma unroll
    for (int kk = 0; kk < CIN; kk += 4) {
        v2f a0 = *(const v2f*)(ap0 + kk);
        v2f a1 = *(const v2f*)(ap1 + kk);
        v2f b0 = *(const v2f*)(bp0 + kk);
        v2f b1 = *(const v2f*)(bp1 + kk);
        acc00 = __builtin_amdgcn_wmma_f32_16x16x4_f32(false, a0, false, b0,
                                                      (short)0, acc00, false, false);
        acc01 = __builtin_amdgcn_wmma_f32_16x16x4_f32(false, a0, false, b1,
                                                      (short)0, acc01, false, false);
        acc10 = __builtin_amdgcn_wmma_f32_16x16x4_f32(false, a1, false, b0,
                                                      (short)0, acc10, false, false);
        acc11 = __builtin_amdgcn_wmma_f32_16x16x4_f32(false, a1, false, b1,
                                                      (short)0, acc11, false, false);
    }
}

__device__ __forceinline__ void td_quad_coords(int& bs0, int& o0, int lid_unused) {
    const int wave = threadIdx.x >> 5;
    const int quad = blockIdx.x * 8 + wave;   // 8 waves/block
    bs0 = (quad >> 3) * 2;                    // pair of (b*S+s) rows-of-16
    o0  = (quad & 7) << 5;                    // pair of channel tiles (32 ch)
}

// ---------------------------------------------------------------------------
// Kernel 2: GEMM pass 1 — per-channel sum / sumsq of h (for BN stats).
// Lanes j and j+16 share channel N=j -> shfl_xor(16) pair reduction.
// ---------------------------------------------------------------------------
__global__ void td_gemm_stats_kernel(const float* __restrict__ x,
                                     const float* __restrict__ W,
                                     const float* __restrict__ bias,
                                     const int* __restrict__ idxg,
                                     float* __restrict__ stats) {
    const int lid = threadIdx.x & 31;
    int bs0, o0;
    td_quad_coords(bs0, o0, lid);

    v8f a00, a01, a10, a11;
    td_gemm_tile22(x, W, idxg, bs0, o0, lid, a00, a01, a10, a11);

    const int  half = lid >> 4;
    const int  m    = lid & 15;
    const float b0  = bias[o0 + m];
    const float b1  = bias[o0 + 16 + m];

    float s1a = 0.f, s2a = 0.f, s1b = 0.f, s2b = 0.f;
#pragma unroll
    for (int i = 0; i < 8; ++i) {
        float h;
        h = a00[i] + b0; s1a += h; s2a += h * h;
        h = a10[i] + b0; s1a += h; s2a += h * h;
        h = a01[i] + b1; s1b += h; s2b += h * h;
        h = a11[i] + b1; s1b += h; s2b += h * h;
    }
    s1a += __shfl_xor(s1a, 16, 32);
    s2a += __shfl_xor(s2a, 16, 32);
    s1b += __shfl_xor(s1b, 16, 32);
    s2b += __shfl_xor(s2b, 16, 32);
    if (half == 0) {
        atomicAdd(&stats[o0 + m], s1a);
        atomicAdd(&stats[COUT + o0 + m], s2a);
        atomicAdd(&stats[o0 + 16 + m], s1b);
        atomicAdd(&stats[COUT + o0 + 16 + m], s2b);
    }
}

// ---------------------------------------------------------------------------
// Kernel 3: fold stats into per-channel scale/shift:
//   scale = gamma * rsqrt(var+eps); shift = beta - mean*scale
// ---------------------------------------------------------------------------
__global__ void td_finalize_kernel(const float* __restrict__ stats,
                                   const float* __restrict__ gamma,
                                   const float* __restrict__ beta,
                                   float* __restrict__ ss) {
    const int o = threadIdx.x;                 // 256 threads
    const float inv  = 1.0f / (float)NROWS;
    const float mean = stats[o] * inv;
    const float var  = stats[COUT + o] * inv - mean * mean;
    const float sc   = gamma[o] * rsqrtf(var + 1e-5f);
    ss[o]        = sc;
    ss[COUT + o] = beta[o] - mean * sc;
}

// ---------------------------------------------------------------------------
// Kernel 4: GEMM pass 2 — fused BN + ReLU + max over K (== max over tile M dim).
// ---------------------------------------------------------------------------
__global__ void td_gemm_out_kernel(const float* __restrict__ x,
                                   const float* __restrict__ W,
                                   const float* __restrict__ bias,
                                   const int* __restrict__ idxg,
                                   const float* __restrict__ ss,
                                   float* __restrict__ out) {
    const int lid = threadIdx.x & 31;
    int bs0, o0;
    td_quad_coords(bs0, o0, lid);

    v8f a00, a01, a10, a11;
    td_gemm_tile22(x, W, idxg, bs0, o0, lid, a00, a01, a10, a11);

    const int  half = lid >> 4;
    const int  m    = lid & 15;
    const float b0  = bias[o0 + m];
    const float b1  = bias[o0 + 16 + m];
    const float sc0 = ss[o0 + m],      sh0 = ss[COUT + o0 + m];
    const float sc1 = ss[o0 + 16 + m], sh1 = ss[COUT + o0 + 16 + m];

    float v00 = 0.f, v01 = 0.f, v10 = 0.f, v11 = 0.f;  // ReLU baked into max
#pragma unroll
    for (int i = 0; i < 8; ++i) {
        v00 = fmaxf(v00, (a00[i] + b0) * sc0 + sh0);
        v10 = fmaxf(v10, (a10[i] + b0) * sc0 + sh0);
        v01 = fmaxf(v01, (a01[i] + b1) * sc1 + sh1);
        v11 = fmaxf(v11, (a11[i] + b1) * sc1 + sh1);
    }
    v00 = fmaxf(v00, __shfl_xor(v00, 16, 32));   // combine M=0..7 with M=8..15
    v10 = fmaxf(v10, __shfl_xor(v10, 16, 32));
    v01 = fmaxf(v01, __shfl_xor(v01, 16, 32));
    v11 = fmaxf(v11, __shfl_xor(v11, 16, 32));
    if (half == 0) {
        out[(size_t)(bs0 + 0) * COUT + o0 + m]      = v00;
        out[(size_t)(bs0 + 1) * COUT + o0 + m]      = v10;
        out[(size_t)(bs0 + 0) * COUT + o0 + 16 + m] = v01;
        out[(size_t)(bs0 + 1) * COUT + o0 + 16 + m] = v11;
    }
}

// ---------------------------------------------------------------------------
extern "C" void kernel_launch(void* const* d_in, const int* in_sizes, int n_in,
                              void* d_out, int out_size, void* d_ws, size_t ws_size,
                              hipStream_t stream) {
    const float* x     = (const float*)d_in[0];
    const float* xyz   = (const float*)d_in[1];
    const float* W     = (const float*)d_in[2];
    const float* b     = (const float*)d_in[3];
    const float* gamma = (const float*)d_in[4];
    const float* beta  = (const float*)d_in[5];
    float* out = (float*)d_out;

    char* ws      = (char*)d_ws;
    int*   idxg   = (int*)ws;                                // 262144 ints (1 MB)
    float* stats  = (float*)(ws + (size_t)NROWS * 4);        // 512 floats
    float* ss     = (float*)(ws + (size_t)NROWS * 4 + 2048); // 512 floats

    hipLaunchKernelGGL(td_zero_kernel, dim3(1), dim3(512), 0, stream, stats);
    hipLaunchKernelGGL(td_knn_kernel, dim3(BB * (SS_ / 256)), dim3(256), 0, stream,
                       xyz, idxg, out);
    hipLaunchKernelGGL(td_gemm_stats_kernel, dim3(NQUADS / 8), dim3(256), 0, stream,
                       x, W, b, idxg, stats);
    hipLaunchKernelGGL(td_finalize_kernel, dim3(1), dim3(256), 0, stream,
                       stats, gamma, beta, ss);
    hipLaunchKernelGGL(td_gemm_out_kernel, dim3(NQUADS / 8), dim3(256), 0, stream,
                       x, W, b, idxg, ss, out);
}